// GeometricRelationalGraphConv_60318520705562
// MI455X (gfx1250) — compile-verified
//
#include <hip/hip_runtime.h>
#include <hip/hip_bf16.h>

// ---- problem constants (match reference) ----
#define NN      20000
#define EE      20000
#define TT      7
#define DIN     128
#define HH      32
#define DOUT    32
#define DE      61
#define BN_EPS  1e-5f
#define NEDGE   (TT*EE)       // 140000
#define NTILE_E (NEDGE/16)    // 8750 (exact)
#define NTILE_N (NN/16)       // 1250 (exact)
// fused contraction: p = k*36 + m ; k in [0,33) (msg_ext), m in [0,36) (hid_ext pad 33->36)
#define KK      33
#define SLOTS   9             // 36/4 WMMA steps per k
#define KSTEPS  (KK*SLOTS)    // 297
#define BPACKSZ (KSTEPS*32*4) // 38016 floats

typedef __attribute__((ext_vector_type(2))) float v2f;
typedef __attribute__((ext_vector_type(8))) float v8f;

__device__ __forceinline__ v8f wmma4(v2f a, v2f b, v8f c) {
  // V_WMMA_F32_16X16X4_F32 : D(16x16,f32) = A(16x4,f32) x B(4x16,f32) + C
  return __builtin_amdgcn_wmma_f32_16x16x4_f32(false, a, false, b, (short)0, c, false, false);
}

// ---- workspace layout (float offsets) ----
#define OFF_BPACK  0u              // 38016   swizzled dynamic-kernel weights
#define OFF_HID    38016u          // 4480000 hid = relu(ef@Wk1+bk1)
#define OFF_MSGP   4518016u        // 4480000 msg_pre (pre-BN)
#define OFF_UPD    8998016u        // 640000  scatter accumulator [N,H]
#define OFF_OPRE   9638016u        // 640000  combine pre-BN [N,DOUT]
#define OFF_MSUM   10278016u       // 224
#define OFF_MSQ    10278240u       // 224
#define OFF_OSUM   10278464u       // 32
#define OFF_OSQ    10278496u       // 32   (MSUM..OSQ = 512 floats, zeroed each call)
#define OFF_MSCALE 10278528u       // 224
#define OFF_MSHIFT 10278752u       // 224
#define OFF_OSCALE 10278976u       // 32
#define OFF_OSHIFT 10279008u       // 32

// ============================================================================
// 0) Pre-swizzle dynamic-kernel weights into per-lane B-fragment order.
//    bpack[(s*32+lane)*4 + q], q = {e,n}: n=q>>1 (N-half), e=q&1 (B vgpr elem)
//    contraction row p = 4s + (lane>>4) + 2e ; col j = n*16 + (lane&15)
//    p -> k = p/36 (msg side), m = p%36 (hid side)
//    value = m<32 ? Wk2[m][k*32+j] : (m==32 ? bk2[k*32+j] : 0)
// ============================================================================
__global__ __launch_bounds__(256) void prep_bpack_kernel(
    const float* __restrict__ Wk2, const float* __restrict__ bk2,
    float* __restrict__ bpack) {
  int idx = blockIdx.x * blockDim.x + threadIdx.x;
  if (idx >= BPACKSZ) return;
  int q    = idx & 3;
  int lane = (idx >> 2) & 31;
  int s    = idx >> 7;
  int p    = s * 4 + (lane >> 4) + ((q & 1) << 1);
  int j    = ((q >> 1) << 4) + (lane & 15);
  int k = p / 36, m = p % 36;
  float v = 0.f;
  if (m < 32)       v = Wk2[m * 1056 + k * 32 + j];
  else if (m == 32) v = bk2[k * 32 + j];
  bpack[idx] = v;
}

// ============================================================================
// 1) hid = relu(edge_features @ Wk1 + bk1)   [140000 x 61] x [61 x 32]
//    One wave per 16-edge tile; K padded 61->64; B pre-swizzled to fragment
//    order in LDS so each fragment is one aligned ds_load_b64.
// ============================================================================
__global__ __launch_bounds__(256) void hid_kernel(
    const float* __restrict__ ef, const float* __restrict__ Wk1,
    const float* __restrict__ bk1, float* __restrict__ hid, int numTiles) {
  __shared__ float sBp[16 * 2 * 32 * 2];   // [step][n][lane][elem]
  for (int i = threadIdx.x; i < 16 * 2 * 32 * 2; i += blockDim.x) {
    int e = i & 1, ln = (i >> 1) & 31, n = (i >> 6) & 1, s = i >> 7;
    int row = 4 * s + (ln >> 4) + 2 * e;
    int col = n * 16 + (ln & 15);
    sBp[i] = (row < DE) ? Wk1[row * 32 + col] : 0.f;
  }
  __syncthreads();
  int wave = threadIdx.x >> 5, lane = threadIdx.x & 31;
  int tile = blockIdx.x * (blockDim.x >> 5) + wave;
  if (tile >= numTiles) return;                    // wave-uniform
  long ebase = (long)tile * 16;
  int row = lane & 15, khalf = (lane >> 4) << 1, j0 = lane & 15;
  const float* erow = ef + (ebase + row) * DE;
  const float2* bf = (const float2*)sBp;
  v8f c0 = {}, c1 = {};
#pragma unroll
  for (int s = 0; s < 16; ++s) {
    int c = s * 4 + khalf;
    v2f a;
    a.x = (c     < DE) ? erow[c]     : 0.f;
    a.y = (c + 1 < DE) ? erow[c + 1] : 0.f;
    float2 b0 = bf[(s * 2 + 0) * 32 + lane];
    float2 b1 = bf[(s * 2 + 1) * 32 + lane];
    v2f b0v = { b0.x, b0.y }, b1v = { b1.x, b1.y };
    c0 = wmma4(a, b0v, c0);
    c1 = wmma4(a, b1v, c1);
  }
  float bb0 = bk1[j0], bb1 = bk1[j0 + 16];
#pragma unroll
  for (int i = 0; i < 8; ++i) {
    int M = ((lane >> 4) << 3) + i;
    hid[(ebase + M) * 32 + j0]      = fmaxf(c0[i] + bb0, 0.f);
    hid[(ebase + M) * 32 + j0 + 16] = fmaxf(c1[i] + bb1, 0.f);
  }
}

// ============================================================================
// 2) msg_pre = gather(nf, src) @ W1 + b1; also accumulate per-(t,h) BN stats.
// ============================================================================
__global__ __launch_bounds__(256) void msg_kernel(
    const float* __restrict__ nf, const int* __restrict__ src,
    const float* __restrict__ W1, const float* __restrict__ b1g,
    float* __restrict__ msgp, float* __restrict__ msum, float* __restrict__ msq,
    int numTiles) {
  __shared__ float sBp[32 * 2 * 32 * 2];   // 16KB, fragment order
  for (int i = threadIdx.x; i < 32 * 2 * 32 * 2; i += blockDim.x) {
    int e = i & 1, ln = (i >> 1) & 31, n = (i >> 6) & 1, s = i >> 7;
    int row = 4 * s + (ln >> 4) + 2 * e;
    int col = n * 16 + (ln & 15);
    sBp[i] = W1[row * 32 + col];
  }
  __syncthreads();
  int wave = threadIdx.x >> 5, lane = threadIdx.x & 31;
  int tile = blockIdx.x * (blockDim.x >> 5) + wave;
  if (tile >= numTiles) return;
  long ebase = (long)tile * 16;
  int t = (int)(ebase / EE);                       // E%16==0: tile within one t
  int row = lane & 15, khalf = (lane >> 4) << 1, j0 = lane & 15;
  int sidx = src[ebase + row];
  const float* xrow = nf + (long)sidx * DIN + khalf;
  const float2* bf = (const float2*)sBp;
  v8f c0 = {}, c1 = {};
#pragma unroll
  for (int s = 0; s < DIN / 4; ++s) {
    v2f a = { xrow[s * 4], xrow[s * 4 + 1] };
    float2 b0 = bf[(s * 2 + 0) * 32 + lane];
    float2 b1 = bf[(s * 2 + 1) * 32 + lane];
    v2f b0v = { b0.x, b0.y }, b1v = { b1.x, b1.y };
    c0 = wmma4(a, b0v, c0);
    c1 = wmma4(a, b1v, c1);
  }
  float bb0 = b1g[j0], bb1 = b1g[j0 + 16];
  float s1a = 0, s2a = 0, s1b = 0, s2b = 0;
#pragma unroll
  for (int i = 0; i < 8; ++i) {
    int M = ((lane >> 4) << 3) + i;
    float va = c0[i] + bb0, vb = c1[i] + bb1;
    msgp[(ebase + M) * 32 + j0]      = va;
    msgp[(ebase + M) * 32 + j0 + 16] = vb;
    s1a += va; s2a += va * va; s1b += vb; s2b += vb * vb;
  }
  atomicAdd(&msum[t * 32 + j0], s1a);      atomicAdd(&msq[t * 32 + j0], s2a);
  atomicAdd(&msum[t * 32 + j0 + 16], s1b); atomicAdd(&msq[t * 32 + j0 + 16], s2b);
}

// ============================================================================
// 3) fold BN stats -> per-(t,h) scale/shift
// ============================================================================
__global__ void mstats_kernel(const float* __restrict__ msum, const float* __restrict__ msq,
                              const float* __restrict__ gm, const float* __restrict__ bm,
                              float* __restrict__ mscale, float* __restrict__ mshift) {
  int i = blockIdx.x * blockDim.x + threadIdx.x;
  if (i >= TT * 32) return;
  int h = i & 31;
  float mean = msum[i] / (float)EE;
  float var  = msq[i] / (float)EE - mean * mean;
  float r    = rsqrtf(var + BN_EPS);
  float sc   = gm[h] * r;
  mscale[i] = sc;
  mshift[i] = bm[h] - mean * sc;
}

// ============================================================================
// 4) Fused dynamic-kernel contraction + shell-weighted scatter.
//    message[e,j] = sum_p hid_ext36[e,p%36] * msg_ext[e,p/36] * B[p,j]
//    k is wave-constant per WMMA step -> A = (ds_load_b64 of 2 hid values) *
//    scalar msg value; B is one coalesced b128 per step.
// ============================================================================
__global__ __launch_bounds__(256) void fused_message_scatter_kernel(
    const float* __restrict__ msgp, const float* __restrict__ hid,
    const float* __restrict__ bpack,
    const float* __restrict__ mscale, const float* __restrict__ mshift,
    const int* __restrict__ dst, const int* __restrict__ shells,
    const float* __restrict__ slw, float* __restrict__ update, int numTiles) {
  __shared__ float sMsg[8][16][36];   // msg_ext: [0]=1, [1..32]=relu(bn(msg))
  __shared__ float sHid[8][16][36];   // hid_ext: [0..31]=hid, [32]=1, [33..35]=0
  int wave = threadIdx.x >> 5, lane = threadIdx.x & 31;
  int tile = blockIdx.x * 8 + wave;
  if (tile >= numTiles) return;                    // wave-uniform
  long ebase = (long)tile * 16;
  int t = (int)(ebase / EE);
  for (int i = lane; i < 512; i += 32) {
    int r = i >> 5, h = i & 31;
    float mp = msgp[(ebase + r) * 32 + h];
    sMsg[wave][r][h + 1] = fmaxf(mp * mscale[t * 32 + h] + mshift[t * 32 + h], 0.f);
    sHid[wave][r][h]     = hid[(ebase + r) * 32 + h];
  }
  if (lane < 16) {
    sMsg[wave][lane][0]  = 1.f;
    sHid[wave][lane][32] = 1.f;
    sHid[wave][lane][33] = 0.f;
    sHid[wave][lane][34] = 0.f;
    sHid[wave][lane][35] = 0.f;
  }
  __builtin_amdgcn_wave_barrier();   // LDS is in-order within a wave; block reordering
  int row = lane & 15, half = lane >> 4;
  const float*  mrow = &sMsg[wave][row][0];
  const float2* hx   = (const float2*)(&sHid[wave][row][half * 2]); // 8B aligned
  v8f c0 = {}, c1 = {};
  const float4* bp = (const float4*)bpack;
#pragma unroll 1
  for (int kk = 0; kk < KK; ++kk) {
    float mval = mrow[kk];
    const float4* bs = bp + (long)kk * SLOTS * 32 + lane;
#pragma unroll
    for (int slot = 0; slot < SLOTS; ++slot) {
      float4 bq = bs[slot * 32];                   // coalesced b128, L2/WGP$-hot
      float2 h2 = hx[slot * 2];                    // hid_ext[m], hid_ext[m+1]
      v2f a  = { h2.x * mval, h2.y * mval };
      v2f b0 = { bq.x, bq.y }, b1v = { bq.z, bq.w };
      c0 = wmma4(a, b0, c0);
      c1 = wmma4(a, b1v, c1);
    }
  }
  int j0 = lane & 15;
#pragma unroll
  for (int i = 0; i < 8; ++i) {
    int M = (half << 3) + i;
    long e = ebase + M;
    int d = dst[e];
    int sh = shells[d];
    int si = min(max(sh - 1, 0), 2);
    float f = 1.f + slw[si];
    atomicAdd(&update[(long)d * 32 + j0],      c0[i] * f);
    atomicAdd(&update[(long)d * 32 + j0 + 16], c1[i] * f);
  }
}

// ============================================================================
// 5) opre = update @ W2 + b2, accumulate per-channel BN stats (over N)
// ============================================================================
__global__ __launch_bounds__(256) void combine_kernel(
    const float* __restrict__ update, const float* __restrict__ W2,
    const float* __restrict__ b2g, float* __restrict__ opre,
    float* __restrict__ osum, float* __restrict__ osq, int numTiles) {
  __shared__ float sBp[8 * 2 * 32 * 2];
  for (int i = threadIdx.x; i < 8 * 2 * 32 * 2; i += blockDim.x) {
    int e = i & 1, ln = (i >> 1) & 31, n = (i >> 6) & 1, s = i >> 7;
    int row = 4 * s + (ln >> 4) + 2 * e;
    int col = n * 16 + (ln & 15);
    sBp[i] = W2[row * 32 + col];
  }
  __syncthreads();
  int wave = threadIdx.x >> 5, lane = threadIdx.x & 31;
  int tile = blockIdx.x * (blockDim.x >> 5) + wave;
  if (tile >= numTiles) return;
  long nbase = (long)tile * 16;
  int row = lane & 15, khalf = (lane >> 4) << 1, j0 = lane & 15;
  const float* urow = update + (nbase + row) * 32 + khalf;
  const float2* bf = (const float2*)sBp;
  v8f c0 = {}, c1 = {};
#pragma unroll
  for (int s = 0; s < 8; ++s) {
    v2f a = { urow[s * 4], urow[s * 4 + 1] };
    float2 b0 = bf[(s * 2 + 0) * 32 + lane];
    float2 b1 = bf[(s * 2 + 1) * 32 + lane];
    v2f b0v = { b0.x, b0.y }, b1v = { b1.x, b1.y };
    c0 = wmma4(a, b0v, c0);
    c1 = wmma4(a, b1v, c1);
  }
  float bb0 = b2g[j0], bb1 = b2g[j0 + 16];
  float s1a = 0, s2a = 0, s1b = 0, s2b = 0;
#pragma unroll
  for (int i = 0; i < 8; ++i) {
    int M = ((lane >> 4) << 3) + i;
    float va = c0[i] + bb0, vb = c1[i] + bb1;
    opre[(nbase + M) * 32 + j0]      = va;
    opre[(nbase + M) * 32 + j0 + 16] = vb;
    s1a += va; s2a += va * va; s1b += vb; s2b += vb * vb;
  }
  atomicAdd(&osum[j0], s1a);      atomicAdd(&osq[j0], s2a);
  atomicAdd(&osum[j0 + 16], s1b); atomicAdd(&osq[j0 + 16], s2b);
}

__global__ void ostats_kernel(const float* __restrict__ osum, const float* __restrict__ osq,
                              const float* __restrict__ go, const float* __restrict__ bo,
                              float* __restrict__ oscale, float* __restrict__ oshift) {
  int j = threadIdx.x;
  if (j >= 32) return;
  float mean = osum[j] / (float)NN;
  float var  = osq[j] / (float)NN - mean * mean;
  float r    = rsqrtf(var + BN_EPS);
  float sc   = go[j] * r;
  oscale[j] = sc;
  oshift[j] = bo[j] - mean * sc;
}

__global__ __launch_bounds__(256) void final_kernel(
    const float* __restrict__ opre, const float* __restrict__ oscale,
    const float* __restrict__ oshift, float* __restrict__ out) {
  int i = blockIdx.x * blockDim.x + threadIdx.x;
  if (i >= NN * DOUT) return;
  int j = i & 31;
  out[i] = fmaxf(opre[i] * oscale[j] + oshift[j], 0.f);
}

// ============================================================================
extern "C" void kernel_launch(void* const* d_in, const int* in_sizes, int n_in,
                              void* d_out, int out_size, void* d_ws, size_t ws_size,
                              hipStream_t stream) {
  const float* nf     = (const float*)d_in[0];
  const float* ef     = (const float*)d_in[1];
  const int*   src    = (const int*)d_in[2];
  const int*   dst    = (const int*)d_in[3];
  const int*   shells = (const int*)d_in[4];
  const float* W1     = (const float*)d_in[5];
  const float* b1     = (const float*)d_in[6];
  const float* Wk1    = (const float*)d_in[7];
  const float* bk1    = (const float*)d_in[8];
  const float* Wk2    = (const float*)d_in[9];
  const float* bk2    = (const float*)d_in[10];
  const float* W2     = (const float*)d_in[11];
  const float* b2     = (const float*)d_in[12];
  const float* gm     = (const float*)d_in[13];
  const float* bm     = (const float*)d_in[14];
  const float* go     = (const float*)d_in[15];
  const float* bo     = (const float*)d_in[16];
  const float* slw    = (const float*)d_in[17];
  float* out = (float*)d_out;
  float* ws  = (float*)d_ws;

  float* bpack  = ws + OFF_BPACK;
  float* hid    = ws + OFF_HID;
  float* msgp   = ws + OFF_MSGP;
  float* upd    = ws + OFF_UPD;
  float* opre   = ws + OFF_OPRE;
  float* msum   = ws + OFF_MSUM;
  float* msq    = ws + OFF_MSQ;
  float* osum   = ws + OFF_OSUM;
  float* osq    = ws + OFF_OSQ;
  float* mscale = ws + OFF_MSCALE;
  float* mshift = ws + OFF_MSHIFT;
  float* oscale = ws + OFF_OSCALE;
  float* oshift = ws + OFF_OSHIFT;

  // zero the accumulators (update + all stat sums) every call
  hipMemsetAsync(upd, 0, (size_t)NN * HH * sizeof(float), stream);
  hipMemsetAsync(msum, 0, 512 * sizeof(float), stream);   // msum,msq,osum,osq contiguous

  prep_bpack_kernel<<<(BPACKSZ + 255) / 256, 256, 0, stream>>>(Wk2, bk2, bpack);

  int tilesE = NTILE_E;                    // 8750
  int blocksE = (tilesE + 7) / 8;          // 1094 (8 waves/block)
  hid_kernel<<<blocksE, 256, 0, stream>>>(ef, Wk1, bk1, hid, tilesE);
  msg_kernel<<<blocksE, 256, 0, stream>>>(nf, src, W1, b1, msgp, msum, msq, tilesE);
  mstats_kernel<<<1, 256, 0, stream>>>(msum, msq, gm, bm, mscale, mshift);
  fused_message_scatter_kernel<<<blocksE, 256, 0, stream>>>(
      msgp, hid, bpack, mscale, mshift, dst, shells, slw, upd, tilesE);

  int tilesN = NTILE_N;                    // 1250
  combine_kernel<<<(tilesN + 7) / 8, 256, 0, stream>>>(upd, W2, b2, opre, osum, osq, tilesN);
  ostats_kernel<<<1, 32, 0, stream>>>(osum, osq, go, bo, oscale, oshift);
  final_kernel<<<(NN * DOUT + 255) / 256, 256, 0, stream>>>(opre, oscale, oshift, out);
}